// DcrROIHead_80745385165042
// MI455X (gfx1250) — compile-verified
//
#include <hip/hip_runtime.h>
#include <stdint.h>

// Problem constants (from reference)
#define B_IMG   16
#define M_GT    256
#define N_PROP  8192
#define NCLS    80
#define KFG     128
#define KBG     384
#define SAMP    512

// ---- CDNA5 async global->LDS path (guarded; falls back to plain LDS fill) ----
#if defined(__HIP_DEVICE_COMPILE__) && __has_builtin(__builtin_amdgcn_global_load_async_to_lds_b128)
#define ASYNC_LDS 1
#endif

#ifdef ASYNC_LDS
// Builtin prototype wants v4i pointers: AS1 (global) source, AS3 (LDS) dest.
typedef int v4i __attribute__((vector_size(16)));
typedef __attribute__((address_space(1))) v4i* g128p;   // 64-bit global pointer
typedef __attribute__((address_space(3))) v4i* l128p;   // 32-bit LDS pointer
#endif

// =====================================================================
// Kernel 1: IoU matching.  grid = (N/256, B), block = 256 (8 wave32).
// GT boxes for the image (4KB) staged in LDS via async global->LDS DMA;
// inner loop over 256 GT boxes is broadcast LDS reads + min/max/fma;
// ratio-compare avoids a divide per iteration (single divide at end).
// =====================================================================
__global__ __launch_bounds__(256) void
iou_match_kernel(const float* __restrict__ gt_boxes,     // [B,M,4]
                 const float* __restrict__ prop_boxes,   // [B,N,4]
                 float*       __restrict__ matched_vals, // [B,N]
                 int*         __restrict__ matched_idxs) // [B,N]
{
    __shared__ float4 gsh[M_GT];     // GT boxes AoS (broadcast reads -> no bank conflicts)
    __shared__ float  garea[M_GT];   // GT areas

    const int b = blockIdx.y;
    const int t = threadIdx.x;
    const int i = blockIdx.x * 256 + t;

    const float4* gsrc = (const float4*)(gt_boxes + (size_t)b * M_GT * 4);
    const float4* psrc = (const float4*)(prop_boxes + (size_t)b * N_PROP * 4);

    // Prefetch this thread's proposal while the LDS fill is in flight
    __builtin_prefetch(psrc + i, 0, 1);   // -> global_prefetch_b8 on gfx1250

#ifdef ASYNC_LDS
    {
        g128p g = (g128p)(uintptr_t)(gsrc + t);
        l128p l = (l128p)(uint32_t)(uintptr_t)(&gsh[t]);
        __builtin_amdgcn_global_load_async_to_lds_b128(g, l, 0, 0);
#if __has_builtin(__builtin_amdgcn_s_wait_asynccnt)
        __builtin_amdgcn_s_wait_asynccnt(0);
#else
        asm volatile("s_wait_asynccnt 0" ::: "memory");
#endif
    }
#else
    gsh[t] = gsrc[t];
#endif
    __syncthreads();

    {
        float4 g = gsh[t];
        garea[t] = (g.z - g.x) * (g.w - g.y);
    }
    __syncthreads();

    const float4 p  = psrc[i];
    const float  ap = (p.z - p.x) * (p.w - p.y);

    // Track best IoU via cross-multiplied ratio compare: inter/uni > bi/bu
    // <=> inter*bu > bi*uni (unions strictly positive here).
    float bi = 0.0f, bu = 1.0f;
    int   bidx = 0;

#pragma unroll 4
    for (int m = 0; m < M_GT; ++m) {
        const float4 q   = gsh[m];
        const float  ltx = fmaxf(q.x, p.x);
        const float  lty = fmaxf(q.y, p.y);
        const float  rbx = fminf(q.z, p.z);
        const float  rby = fminf(q.w, p.w);
        const float  w   = fmaxf(rbx - ltx, 0.0f);
        const float  h   = fmaxf(rby - lty, 0.0f);
        const float  inter = w * h;
        const float  uni   = garea[m] + ap - inter;
        const bool   better = inter * bu > bi * uni;  // strict > keeps first-max (argmax semantics)
        bi   = better ? inter : bi;
        bu   = better ? uni   : bu;
        bidx = better ? m     : bidx;
    }

    const size_t o = (size_t)b * N_PROP + i;
    matched_vals[o] = (bi > 0.0f) ? (bi / bu) : 0.0f;
    matched_idxs[o] = bidx;
}

// =====================================================================
// Kernel 2: exact top-k sampling. 1 block per image, 256 threads.
// 64-bit key per proposal in LDS (66KB — fine inside CDNA5's 320KB WGP
// LDS).  key = ((pri_bits+1) << 32) | (N - i) when masked,
//        key =                        (N - i) when unmasked.
// This reproduces lax.top_k exactly: descending priority, lower index
// wins ties, and the NEG (unmasked) tail comes out in ascending index
// order.  k block-wide u64-max tournaments; winner index is embedded in
// the key so removal is a single LDS store.
// =====================================================================
__device__ __forceinline__ int assigned_class(const int* gc, const float* mv,
                                              const int* mi, int i)
{
    int c = gc[mi[i]];
    c = (c == NCLS) ? -1 : c;              // ignored-class branch (inert here)
    c = (mv[i] >= 0.5f) ? c : NCLS;        // background if unmatched
    return c;
}

__global__ __launch_bounds__(256) void
sample_kernel(const int*   __restrict__ gt_classes,   // [B,M]
              const float* __restrict__ rand_pri,     // [B,N]
              const float* __restrict__ matched_vals, // [B,N] (from kernel 1)
              const int*   __restrict__ matched_idxs, // [B,N]
              int* __restrict__ s_idx,                // [B,512]
              int* __restrict__ s_cls,                // [B,512]
              int* __restrict__ s_gt)                 // [B,512]
{
    __shared__ unsigned long long keys[N_PROP];  // 64KB
    __shared__ unsigned long long red[256];      //  2KB

    const int b = blockIdx.x;
    const int t = threadIdx.x;

    const float* mv = matched_vals + (size_t)b * N_PROP;
    const int*   mi = matched_idxs + (size_t)b * N_PROP;
    const int*   gc = gt_classes   + (size_t)b * M_GT;
    const float* pr = rand_pri     + (size_t)b * N_PROP;

    for (int phase = 0; phase < 2; ++phase) {            // 0 = fg, 1 = bg
        // ---- build keys (bg phase rebuilds fresh: fg/bg top_k are independent) ----
        for (int i = t; i < N_PROP; i += 256) {
            const int  c   = assigned_class(gc, mv, mi, i);
            const bool sel = (phase == 0) ? (c >= 0 && c != NCLS) : (c == NCLS);
            unsigned long long k = (unsigned long long)(N_PROP - i);   // 1..8192, never 0
            if (sel) {
                const unsigned pb = __float_as_uint(pr[i]) + 1u;       // [0,1) bits + 1 -> >=1
                k |= ((unsigned long long)pb) << 32;
            }
            keys[i] = k;
        }
        __syncthreads();

        const int kcnt = (phase == 0) ? KFG : KBG;
        const int base = (phase == 0) ? 0   : KFG;

        for (int slot = 0; slot < kcnt; ++slot) {
            // block-wide u64 max
            unsigned long long best = 0ull;
            for (int i = t; i < N_PROP; i += 256) {
                const unsigned long long k = keys[i];
                best = (k > best) ? k : best;
            }
            red[t] = best;
            __syncthreads();
            for (int s = 128; s > 0; s >>= 1) {
                if (t < s) {
                    const unsigned long long o = red[t + s];
                    if (o > red[t]) red[t] = o;
                }
                __syncthreads();
            }

            if (t == 0) {
                const unsigned long long win = red[0];
                const int  idx   = N_PROP - (int)(win & 0xFFFFFFFFull);
                const bool valid = (win >> 32) != 0ull;   // == (top > NEG)
                keys[idx] = 0ull;                          // remove from tournament
                const int off = b * SAMP + base + slot;
                int sc = -1, sg = -1;
                if (valid) {
                    sc = assigned_class(gc, mv, mi, idx);
                    sg = mi[idx];
                }
                s_idx[off] = idx;
                s_cls[off] = sc;
                s_gt[off]  = sg;
            }
            __syncthreads();
        }
        __syncthreads();
    }
}

// =====================================================================
// Launch.  Output layout (flat concat, 4-byte elems; ints stored as
// int32 bit patterns):
//   [0,                B*N)   matched_vals  (f32)
//   [B*N,            2*B*N)   matched_idxs  (i32)
//   [2*B*N,      +B*512)      sampled_idxs  (i32)
//   [...,        +B*512)      sampled_classes (i32)
//   [...,        +B*512)      sampled_gt    (i32)
// =====================================================================
extern "C" void kernel_launch(void* const* d_in, const int* in_sizes, int n_in,
                              void* d_out, int out_size, void* d_ws, size_t ws_size,
                              hipStream_t stream)
{
    (void)in_sizes; (void)n_in; (void)out_size; (void)d_ws; (void)ws_size;

    const float* gt_boxes   = (const float*)d_in[0];
    const int*   gt_classes = (const int*)  d_in[1];
    const float* prop_boxes = (const float*)d_in[2];
    const float* rand_pri   = (const float*)d_in[3];

    float* out          = (float*)d_out;
    float* matched_vals = out;
    int*   matched_idxs = (int*)(out + (size_t)B_IMG * N_PROP);
    int*   s_idx        = (int*)(out + 2 * (size_t)B_IMG * N_PROP);
    int*   s_cls        = s_idx + B_IMG * SAMP;
    int*   s_gt         = s_cls + B_IMG * SAMP;

    dim3 g1(N_PROP / 256, B_IMG);
    iou_match_kernel<<<g1, 256, 0, stream>>>(gt_boxes, prop_boxes,
                                             matched_vals, matched_idxs);

    sample_kernel<<<B_IMG, 256, 0, stream>>>(gt_classes, rand_pri,
                                             matched_vals, matched_idxs,
                                             s_idx, s_cls, s_gt);
}